// MyAttention_11416023072831
// MI455X (gfx1250) — compile-verified
//
#include <hip/hip_runtime.h>

typedef __attribute__((ext_vector_type(2))) float v2f;
typedef __attribute__((ext_vector_type(8))) float v8f;
typedef __attribute__((ext_vector_type(4))) unsigned int u32x4;
typedef __attribute__((ext_vector_type(4))) int i32x4;
typedef __attribute__((ext_vector_type(8))) int i32x8;

#ifndef __has_builtin
#define __has_builtin(x) 0
#endif
#if __has_builtin(__builtin_amdgcn_tensor_load_to_lds)
#define HAVE_TDM 1
#else
#define HAVE_TDM 0
#endif
#if __has_include(<hip/amd_detail/amd_gfx1250_TDM.h>)
#define TDM_6ARG 1
#else
#define TDM_6ARG 0
#endif

namespace {

constexpr int   kB     = 512;
constexpr int   kC     = 768;
constexpr int   kH     = 12;
constexpr int   kHD    = 64;
constexpr int   kL     = 256;
constexpr int   kNp    = 255;
constexpr float kScale = 0.125f;

// LDS row stride for staged xp chunks: 768 + 3*4 pad DWORDs (TDM pads 4 DWORDs
// every 256).  780 % 64 == 12 -> both LDS gather patterns are bank-conflict-free.
constexpr int kStride  = 780;
constexpr int kChunk   = 16;               // kv rows per chunk
constexpr int kNChunks = kL / kChunk;      // 16

// D = A(16x4 f32) * B(4x16 f32) + C, full-precision WMMA.
__device__ __forceinline__ v8f wmma4(v2f a, v2f b, v8f c) {
  return __builtin_amdgcn_wmma_f32_16x16x4_f32(false, a, false, b, (short)0, c,
                                               false, false);
}

#if HAVE_TDM
// Issue one TDM 2-D tile load: nrows x 768 f32 from global (row pitch 768) into
// LDS at lds_byte, padding +4 DWORDs every 256 (row stride becomes 780 floats).
__device__ __forceinline__ void tdm_load_rows(unsigned int lds_byte,
                                              const float* gaddr, int nrows) {
  unsigned long long ga = (unsigned long long)(size_t)gaddr;
  u32x4 g0;
  g0[0] = 1u;                                     // count=1, user descriptor
  g0[1] = lds_byte;                               // lds_addr
  g0[2] = (unsigned int)ga;                       // global_addr[31:0]
  g0[3] = (unsigned int)((ga >> 32) & 0x01FFFFFFu) | (2u << 30);  // addr hi | type=2
  i32x8 g1;
  g1[0] = (int)0x07D20000u;  // data_size=4B, pad_enable, interval=256dw, amount=4dw
  g1[1] = (int)(768u << 16); // tensor_dim0 = 768 (elements)
  g1[2] = (int)((unsigned int)nrows << 16);  // tensor_dim1
  g1[3] = (int)(768u << 16); // tile_dim0 = 768
  g1[4] = nrows;             // tile_dim1
  g1[5] = 768;               // tensor_dim0_stride
  g1[6] = 0;
  g1[7] = 0;
  i32x4 z4 = {0, 0, 0, 0};
#if TDM_6ARG
  i32x8 z8 = {0, 0, 0, 0, 0, 0, 0, 0};
  __builtin_amdgcn_tensor_load_to_lds(g0, g1, z4, z4, z8, 0);
#else
  __builtin_amdgcn_tensor_load_to_lds(g0, g1, z4, z4, 0);
#endif
}
#endif  // HAVE_TDM

// ---------------------------------------------------------------------------
// K1: Q[b,c] = scale * (x[b,:] @ wq[:,c] + bq[c]).  Wave per 16x16 tile.
// ---------------------------------------------------------------------------
__global__ __launch_bounds__(32) void k_q(const float* __restrict__ x,
                                          const float* __restrict__ wq,
                                          const float* __restrict__ bq,
                                          float* __restrict__ Q) {
  const int tile = blockIdx.x;
  const int mt = tile / 48, nt = tile % 48;
  const int lane = threadIdx.x & 31;
  const int lg = lane >> 4, ln = lane & 15;

  const float* arow = x + (size_t)(mt * 16 + ln) * kC;
  const int nc = nt * 16 + ln;

  v8f acc;
#pragma unroll
  for (int r = 0; r < 8; ++r) acc[r] = 0.f;

  for (int k0 = 0; k0 < kC; k0 += 4) {
    const int kr = k0 + lg * 2;
    v2f a = *(const v2f*)(arow + kr);
    v2f b;
    b.x = wq[(size_t)kr * kC + nc];
    b.y = wq[(size_t)(kr + 1) * kC + nc];
    acc = wmma4(a, b, acc);
  }

  const float bqv = bq[nc];
#pragma unroll
  for (int r = 0; r < 8; ++r) {
    const int m = mt * 16 + r + 8 * lg;
    Q[(size_t)m * kC + nc] = (acc[r] + bqv) * kScale;
  }
}

// ---------------------------------------------------------------------------
// K1b: qb[b,h] = sum_d Q[b,h*64+d] * bk[h*64+d]
// ---------------------------------------------------------------------------
__global__ __launch_bounds__(256) void k_qb(const float* __restrict__ Q,
                                            const float* __restrict__ bk,
                                            float* __restrict__ qb) {
  const int i = blockIdx.x * blockDim.x + threadIdx.x;
  if (i >= kB * kH) return;
  const int b = i / kH, h = i % kH;
  const float* qp = Q + (size_t)b * kC + h * kHD;
  const float* bp = bk + h * kHD;
  float s = 0.f;
#pragma unroll 8
  for (int d = 0; d < kHD; ++d) s += qp[d] * bp[d];
  qb[i] = s;
}

// ---------------------------------------------------------------------------
// K2: T[b,h,c] = sum_d Q[b,h*64+d] * wk[c,h*64+d]   (per-head, B transposed)
// ---------------------------------------------------------------------------
__global__ __launch_bounds__(32) void k_t(const float* __restrict__ Q,
                                          const float* __restrict__ wk,
                                          float* __restrict__ T) {
  const int tile = blockIdx.x;
  const int h = tile / (32 * 48);
  const int rem = tile % (32 * 48);
  const int bt = rem / 48, ct = rem % 48;
  const int lane = threadIdx.x & 31;
  const int lg = lane >> 4, ln = lane & 15;

  const float* arow = Q + (size_t)(bt * 16 + ln) * kC + h * kHD;
  const float* brow = wk + (size_t)(ct * 16 + ln) * kC + h * kHD;

  v8f acc;
#pragma unroll
  for (int r = 0; r < 8; ++r) acc[r] = 0.f;

#pragma unroll
  for (int k0 = 0; k0 < kHD; k0 += 4) {
    v2f a = *(const v2f*)(arow + k0 + lg * 2);
    v2f b = *(const v2f*)(brow + k0 + lg * 2);
    acc = wmma4(a, b, acc);
  }

  const int c = ct * 16 + ln;
#pragma unroll
  for (int r = 0; r < 8; ++r) {
    const int bb = bt * 16 + r + 8 * lg;
    T[(size_t)bb * (kH * kC) + (size_t)h * kC + c] = acc[r];
  }
}

// ---------------------------------------------------------------------------
// K3 (fused): per batch, single pass over xp with online softmax.
//   raw_attn[b,h,l] = xp[l,:].T_b[h,:] + qb[b,h]      -> written to d_out
//   Wbar[b,h,c]     = scale * softmax(raw)[h,:] @ xp  -> workspace
// One 256-thread block (8 waves) per batch.  xp chunks of 16 kv rows are
// DMA'd into double-buffered LDS by the Tensor Data Mover (TENSORcnt) while
// the previous chunk is consumed by two WMMA stages.
// ---------------------------------------------------------------------------
__global__ __launch_bounds__(256) void k_attn(const float* __restrict__ x,
                                              const float* __restrict__ prompts,
                                              const float* __restrict__ T,
                                              const float* __restrict__ qb,
                                              float* __restrict__ attn_out,
                                              float* __restrict__ Wbar) {
#if HAVE_TDM
  constexpr int kNBuf = 2;
#else
  constexpr int kNBuf = 1;
#endif
  __shared__ __align__(16) float sxp[kNBuf][kChunk * kStride];  // staged xp chunks
  __shared__ __align__(16) float spart[8 * 256];                // per-wave S partials
  __shared__ __align__(16) float sp16[16 * 20];                 // chunk probs (padded)
  __shared__ float sS[kH * 16];                                 // reduced chunk logits
  __shared__ float smax[16], ssum[16], sfac[16];

  const int b = blockIdx.x;
  const int t = threadIdx.x;
  const int lane = t & 31, w = t >> 5;
  const int lg = lane >> 4, ln = lane & 15;

  // ---- prologue: init softmax state, pad rows of sp16, kick off chunk 0 ----
  if (t < 16) {
    smax[t] = -3.0e38f;
    ssum[t] = 0.f;
    sfac[t] = 1.f;
  }
  for (int idx = t; idx < 16 * 20; idx += 256) sp16[idx] = 0.f;

#if HAVE_TDM
  if (w == 0) {
    // chunk 0: rows 1..15 come from prompts[0..14]; row 0 (x) loaded below.
    tdm_load_rows((unsigned int)(size_t)&sxp[0][0] + kStride * 4,
                  prompts + (size_t)b * kNp * kC, kChunk - 1);
  }
  if (w == 1) {
    // x row -> sxp[0] row 0, applying the +4-DWORD-per-256 pad pattern.
    const float* xr = x + (size_t)b * kC;
#pragma unroll
    for (int i = 0; i < 6; ++i) {
      const int c = lane * 24 + i * 4;
      *(float4*)(&sxp[0][c + 4 * (c >> 8)]) = *(const float4*)(xr + c);
    }
  }
  if (w == 0) __builtin_amdgcn_s_wait_tensorcnt(0);
#endif
  __syncthreads();

  // persistent Wbar accumulators: wave w owns c-tiles w*6 .. w*6+5
  v8f acc[6];
#pragma unroll
  for (int j = 0; j < 6; ++j)
#pragma unroll
    for (int r = 0; r < 8; ++r) acc[j][r] = 0.f;

  const float* brow = T + (size_t)b * (kH * kC) + (size_t)(ln % kH) * kC;

  for (int ch = 0; ch < kNChunks; ++ch) {
#if HAVE_TDM
    const float* bufp = &sxp[ch & 1][0];
    if (w == 0 && ch + 1 < kNChunks) {
      // prefetch next chunk into the other buffer while we compute
      tdm_load_rows((unsigned int)(size_t)&sxp[(ch + 1) & 1][0],
                    prompts + ((size_t)b * kNp + (ch + 1) * kChunk - 1) * kC,
                    kChunk);
    }
#else
    const float* bufp = &sxp[0][0];
    {  // cooperative staging fallback: 16 rows x 768 floats, padded stride
      const int r = t >> 4, sub = t & 15;
      const int l = ch * kChunk + r;
      const float* rowp = (l == 0)
                              ? (x + (size_t)b * kC)
                              : (prompts + ((size_t)b * kNp + l - 1) * kC);
#pragma unroll
      for (int i = 0; i < 12; ++i) {
        const int c = sub * 48 + i * 4;
        *(float4*)(&sxp[0][r * kStride + c + 4 * (c >> 8)]) =
            *(const float4*)(rowp + c);
      }
    }
    __syncthreads();
#endif

    // ---- phase A: partial logits S[l,h], K split across 8 waves ----
    {
      v8f sacc;
#pragma unroll
      for (int r = 0; r < 8; ++r) sacc[r] = 0.f;
      const int cbase = w * 96;
#pragma unroll
      for (int k0 = 0; k0 < 96; k0 += 4) {
        const int cg = cbase + k0 + lg * 2;
        v2f a = *(const v2f*)(bufp + ln * kStride + cg + 4 * (cg >> 8));
        v2f bb = *(const v2f*)(brow + cg);
        sacc = wmma4(a, bb, sacc);
      }
      float4 p0, p1;
      p0.x = sacc[0]; p0.y = sacc[1]; p0.z = sacc[2]; p0.w = sacc[3];
      p1.x = sacc[4]; p1.y = sacc[5]; p1.z = sacc[6]; p1.w = sacc[7];
      float* dst = &spart[(w * 32 + lane) * 8];
      *(float4*)dst = p0;
      *(float4*)(dst + 4) = p1;
    }
    __syncthreads();

    // ---- reduce partials, emit raw logits, stash for softmax ----
    {
      float red = 0.f;
#pragma unroll
      for (int ww = 0; ww < 8; ++ww) red += spart[ww * 256 + t];
      const int hh = (t >> 3) & 15;
      const int mm = (t & 7) + 8 * (t >> 7);
      if (hh < kH) {
        red += qb[b * kH + hh];
        attn_out[((size_t)b * kH + hh) * kL + ch * kChunk + mm] = red;
        sS[hh * 16 + mm] = red;
      }
    }
    __syncthreads();

    // ---- online softmax update (wave 0, one lane per head) ----
    if (w == 0 && lane < kH) {
      const int h = lane;
      const float mo = smax[h];
      float cm = -3.0e38f;
#pragma unroll
      for (int j = 0; j < 16; ++j) cm = fmaxf(cm, sS[h * 16 + j]);
      const float mn = fmaxf(mo, cm);
      const float f = __expf(mo - mn);
      float sum = 0.f;
#pragma unroll
      for (int j = 0; j < 16; ++j) {
        const float p = __expf(sS[h * 16 + j] - mn);
        sp16[h * 20 + j] = p;
        sum += p;
      }
      ssum[h] = ssum[h] * f + sum;
      smax[h] = mn;
      sfac[h] = f;
    }
    __syncthreads();

    // ---- phase B: rescale running Wbar and accumulate this chunk ----
    {
      const float f = sfac[ln];
#pragma unroll
      for (int j = 0; j < 6; ++j)
#pragma unroll
        for (int r = 0; r < 8; ++r) acc[j][r] *= f;
#pragma unroll
      for (int ks = 0; ks < 4; ++ks) {
        const int lrow = ks * 4 + lg * 2;
        v2f bb = *(const v2f*)(&sp16[ln * 20 + lrow]);
#pragma unroll
        for (int j = 0; j < 6; ++j) {
          const int c0 = (w * 6 + j) * 16;
          const int ca = c0 + 4 * (c0 >> 8) + ln;
          v2f a;
          a.x = bufp[lrow * kStride + ca];
          a.y = bufp[(lrow + 1) * kStride + ca];
          acc[j] = wmma4(a, bb, acc[j]);
        }
      }
    }
#if HAVE_TDM
    if (w == 0) __builtin_amdgcn_s_wait_tensorcnt(0);
#endif
    __syncthreads();
  }

  // ---- epilogue: normalize (scale / denom) and store Wbar ----
  if (ln < kH) {
    const float mult = kScale / ssum[ln];
#pragma unroll
    for (int j = 0; j < 6; ++j) {
      const int c0 = (w * 6 + j) * 16;
#pragma unroll
      for (int r = 0; r < 8; ++r) {
        const int c = c0 + r + 8 * lg;
        Wbar[(size_t)b * (kH * kC) + (size_t)ln * kC + c] = acc[j][r] * mult;
      }
    }
  }
}

// ---------------------------------------------------------------------------
// K5: ctx[b,h*64+d] = sum_c Wbar[b,h,c] * wv[c,h*64+d] + scale*bv[h*64+d]
// ---------------------------------------------------------------------------
__global__ __launch_bounds__(32) void k_ctx(const float* __restrict__ Wbar,
                                            const float* __restrict__ wv,
                                            const float* __restrict__ bv,
                                            float* __restrict__ ctx) {
  const int tile = blockIdx.x;
  const int h = tile / (32 * 4);
  const int rem = tile % (32 * 4);
  const int bt = rem / 4, dt = rem % 4;
  const int lane = threadIdx.x & 31;
  const int lg = lane >> 4, ln = lane & 15;

  const float* arow = Wbar + (size_t)(bt * 16 + ln) * (kH * kC) + (size_t)h * kC;
  const int col = h * kHD + dt * 16 + ln;

  v8f acc;
#pragma unroll
  for (int r = 0; r < 8; ++r) acc[r] = 0.f;

  for (int k0 = 0; k0 < kC; k0 += 4) {
    const int kr = k0 + lg * 2;
    v2f a = *(const v2f*)(arow + kr);
    v2f b;
    b.x = wv[(size_t)kr * kC + col];
    b.y = wv[(size_t)(kr + 1) * kC + col];
    acc = wmma4(a, b, acc);
  }

  const float add = kScale * bv[col];
#pragma unroll
  for (int r = 0; r < 8; ++r) {
    const int bb = bt * 16 + r + 8 * lg;
    ctx[(size_t)bb * kC + col] = acc[r] + add;
  }
}

// ---------------------------------------------------------------------------
// K6: x_cls[b,c] = ctx[b,:] @ wp[:,c] + bp[c] + x[b,c]
// ---------------------------------------------------------------------------
__global__ __launch_bounds__(32) void k_proj(const float* __restrict__ ctx,
                                             const float* __restrict__ wp,
                                             const float* __restrict__ bp,
                                             const float* __restrict__ x,
                                             float* __restrict__ out) {
  const int tile = blockIdx.x;
  const int mt = tile / 48, nt = tile % 48;
  const int lane = threadIdx.x & 31;
  const int lg = lane >> 4, ln = lane & 15;

  const float* arow = ctx + (size_t)(mt * 16 + ln) * kC;
  const int nc = nt * 16 + ln;

  v8f acc;
#pragma unroll
  for (int r = 0; r < 8; ++r) acc[r] = 0.f;

  for (int k0 = 0; k0 < kC; k0 += 4) {
    const int kr = k0 + lg * 2;
    v2f a = *(const v2f*)(arow + kr);
    v2f b;
    b.x = wp[(size_t)kr * kC + nc];
    b.y = wp[(size_t)(kr + 1) * kC + nc];
    acc = wmma4(a, b, acc);
  }

  const float bpv = bp[nc];
#pragma unroll
  for (int r = 0; r < 8; ++r) {
    const int m = mt * 16 + r + 8 * lg;
    out[(size_t)m * kC + nc] = acc[r] + bpv + x[(size_t)m * kC + nc];
  }
}

}  // namespace

extern "C" void kernel_launch(void* const* d_in, const int* in_sizes, int n_in,
                              void* d_out, int out_size, void* d_ws, size_t ws_size,
                              hipStream_t stream) {
  (void)in_sizes; (void)n_in; (void)out_size; (void)ws_size;

  const float* x       = (const float*)d_in[0];
  const float* prompts = (const float*)d_in[1];
  const float* wq      = (const float*)d_in[2];
  const float* bq      = (const float*)d_in[3];
  const float* wk      = (const float*)d_in[4];
  const float* bk      = (const float*)d_in[5];
  const float* wv      = (const float*)d_in[6];
  const float* bv      = (const float*)d_in[7];
  const float* wp      = (const float*)d_in[8];
  const float* bp      = (const float*)d_in[9];

  float* out      = (float*)d_out;
  float* attn_out = out + (size_t)kB * kC;  // raw_attn section of d_out

  float* ws  = (float*)d_ws;
  float* Q   = ws;                       // 512*768
  float* qb  = Q + (size_t)kB * kC;      // 512*12
  float* T   = qb + (size_t)kB * kH;     // 512*12*768 (Wbar aliases T, per-batch)
  float* ctx = T + (size_t)kB * kH * kC; // 512*768   (~22 MB total)

  k_q   <<<dim3(32 * 48),      dim3(32),  0, stream>>>(x, wq, bq, Q);
  k_qb  <<<dim3(24),           dim3(256), 0, stream>>>(Q, bk, qb);
  k_t   <<<dim3(12 * 32 * 48), dim3(32),  0, stream>>>(Q, wk, T);
  k_attn<<<dim3(kB),           dim3(256), 0, stream>>>(x, prompts, T, qb,
                                                       attn_out, T /*Wbar*/);
  k_ctx <<<dim3(12 * 32 * 4),  dim3(32),  0, stream>>>(T, wv, bv, ctx);
  k_proj<<<dim3(32 * 48),      dim3(32),  0, stream>>>(ctx, wp, bp, x, out);
}